// GetCost_initvolume_43069932044810
// MI455X (gfx1250) — compile-verified
//
#include <hip/hip_runtime.h>

#define B_  4
#define V_  5
#define C_  32
#define H_  256
#define W_  320
#define HW_ (H_*W_)
#define G_  8
#define D_  4   // CostNum

typedef __attribute__((ext_vector_type(2))) float v2f;
typedef __attribute__((ext_vector_type(8))) float v8f;

// ---------------------------------------------------------------------------
// Projection precompute: one wave per batch.
// Builds P_v = K_v @ E_v (4x4, row3 = [0,0,0,1]), inverts P_0 (affine),
// then ONE v_wmma_f32_16x16x4_f32 computes proj_v = P_v @ inv(P_0) for all
// four source views at once (A rows 4*vi..4*vi+3 = P_{vi+1}, B = inv(P_0)).
// The 16x16 D tile is dumped to ws[b*256 + m*16 + n].
// ---------------------------------------------------------------------------
__global__ __launch_bounds__(32) void proj_wmma_kernel(
    const float* __restrict__ proj, float* __restrict__ ws)
{
    const int b    = blockIdx.x;
    const int lane = threadIdx.x;
    __shared__ float Pm[V_][16];
    __shared__ float invR[16];

    // Pm[v] = proj_new(b, v): top 3x4 = K[:3,:3] @ E[:3,:4], row 3 = E row 3
    for (int e = lane; e < V_ * 16; e += 32) {
        int v = e >> 4, rc = e & 15, r = rc >> 2, c = rc & 3;
        const float* E = proj + (((b * V_ + v) * 2 + 0) << 4);
        const float* K = proj + (((b * V_ + v) * 2 + 1) << 4);
        float val;
        if (r < 3)
            val = K[r*4+0]*E[0+c] + K[r*4+1]*E[4+c] + K[r*4+2]*E[8+c];
        else
            val = E[12 + c];
        Pm[v][rc] = val;
    }
    __syncthreads();

    if (lane == 0) {
        // affine inverse of Pm[0] = [[A, t],[0,1]] -> [[A^-1, -A^-1 t],[0,1]]
        const float a00=Pm[0][0], a01=Pm[0][1], a02=Pm[0][2],  t0=Pm[0][3];
        const float a10=Pm[0][4], a11=Pm[0][5], a12=Pm[0][6],  t1=Pm[0][7];
        const float a20=Pm[0][8], a21=Pm[0][9], a22=Pm[0][10], t2=Pm[0][11];
        float det = a00*(a11*a22 - a12*a21) - a01*(a10*a22 - a12*a20)
                  + a02*(a10*a21 - a11*a20);
        float id = 1.0f / det;
        float i00=(a11*a22-a12*a21)*id, i01=(a02*a21-a01*a22)*id, i02=(a01*a12-a02*a11)*id;
        float i10=(a12*a20-a10*a22)*id, i11=(a00*a22-a02*a20)*id, i12=(a02*a10-a00*a12)*id;
        float i20=(a10*a21-a11*a20)*id, i21=(a01*a20-a00*a21)*id, i22=(a00*a11-a01*a10)*id;
        invR[0]=i00;  invR[1]=i01;  invR[2]=i02;  invR[3]=-(i00*t0+i01*t1+i02*t2);
        invR[4]=i10;  invR[5]=i11;  invR[6]=i12;  invR[7]=-(i10*t0+i11*t1+i12*t2);
        invR[8]=i20;  invR[9]=i21;  invR[10]=i22; invR[11]=-(i20*t0+i21*t1+i22*t2);
        invR[12]=0.f; invR[13]=0.f; invR[14]=0.f; invR[15]=1.f;
    }
    __syncthreads();

    // A-matrix (16x4 f32, 2 VGPRs): lanes 0-15 hold K=0,1; lanes 16-31 K=2,3
    const int m  = lane & 15;
    const int kb = (lane < 16) ? 0 : 2;
    const int vv = 1 + (m >> 2);
    const int rr = m & 3;
    v2f a;
    a.x = Pm[vv][rr*4 + kb + 0];
    a.y = Pm[vv][rr*4 + kb + 1];

    // B-matrix (4x16 f32, 2 VGPRs): VGPR0 rows K=0/2, VGPR1 rows K=1/3
    const int n = lane & 15;
    v2f bm;
    bm.x = (n < 4) ? invR[(kb + 0)*4 + n] : 0.0f;
    bm.y = (n < 4) ? invR[(kb + 1)*4 + n] : 0.0f;

    v8f c = {};
    c = __builtin_amdgcn_wmma_f32_16x16x4_f32(
            /*neg_a=*/false, a, /*neg_b=*/false, bm,
            /*c_mod=*/(short)0, c, /*reuse_a=*/false, /*reuse_b=*/false);

    // D layout: VGPR r holds row M = r (lanes 0-15) / r+8 (lanes 16-31), col = lane%16
    #pragma unroll
    for (int r2 = 0; r2 < 8; ++r2) {
        int mrow = r2 + ((lane < 16) ? 0 : 8);
        ws[b*256 + mrow*16 + n] = c[r2];
    }
}

// ---------------------------------------------------------------------------
// Feature transpose: (B*V, C, HW) -> (B*V, HW, C). Coalesced strided reads,
// each lane writes its own contiguous 128B channel block (full cachelines).
// ---------------------------------------------------------------------------
__global__ __launch_bounds__(256) void transpose_kernel(
    const float* __restrict__ f, float* __restrict__ ft)
{
    long t = (long)blockIdx.x * blockDim.x + threadIdx.x;
    if (t >= (long)B_ * V_ * HW_) return;
    long bv = t / HW_;
    long hw = t - bv * HW_;
    const float* src = f + bv * (long)C_ * HW_ + hw;
    float4* dst = (float4*)(ft + (bv * (long)HW_ + hw) * C_);
    #pragma unroll
    for (int j = 0; j < 8; ++j) {
        float4 v;
        v.x = src[(4*j + 0) * (long)HW_];
        v.y = src[(4*j + 1) * (long)HW_];
        v.z = src[(4*j + 2) * (long)HW_];
        v.w = src[(4*j + 3) * (long)HW_];
        dst[j] = v;
    }
}

// ---------------------------------------------------------------------------
// Main cost-volume kernel. One thread per output pixel. TR selects the
// channel-last (transposed) fast path (float4 gathers) vs native layout.
// ---------------------------------------------------------------------------
template<bool TR>
__global__ __launch_bounds__(256) void getcost_kernel(
    const float* __restrict__ dvals, const float* __restrict__ dint,
    const float* __restrict__ vwts,  const float* __restrict__ feat,
    const float* __restrict__ projws,
    float* __restrict__ simOut, float* __restrict__ sampOut)
{
    const int pix = blockIdx.x * blockDim.x + threadIdx.x;
    if (pix >= B_ * HW_) return;
    const int b  = pix / HW_;
    const int hw = pix - b * HW_;
    const int h  = hw / W_;
    const int w  = hw - h * W_;

    // inverse-depth range samples (also an output)
    const float inv_d = 1.0f / dvals[pix];
    const float itv   = dint[pix];
    const float dmin  = inv_d - 2.0f * itv;
    const float step  = (4.0f * itv) * (1.0f / 3.0f);
    float depth[D_];
    #pragma unroll
    for (int d = 0; d < D_; ++d) {
        depth[d] = 1.0f / (dmin + (float)d * step);
        sampOut[((long)b * D_ + d) * HW_ + hw] = depth[d];
    }

    // reference features (32 channels) in registers
    float refa[C_];
    if (TR) {
        const float4* rp = (const float4*)(feat + ((long)(b * V_) * HW_ + hw) * C_);
        #pragma unroll
        for (int g = 0; g < 8; ++g) {
            float4 rv = rp[g];
            refa[4*g+0] = rv.x; refa[4*g+1] = rv.y;
            refa[4*g+2] = rv.z; refa[4*g+3] = rv.w;
        }
    } else {
        const float* rp = feat + (long)(b * V_) * C_ * HW_ + hw;
        #pragma unroll
        for (int cc = 0; cc < C_; ++cc) refa[cc] = rp[(long)cc * HW_];
    }

    float acc[G_][D_];
    #pragma unroll
    for (int g = 0; g < G_; ++g)
        #pragma unroll
        for (int d = 0; d < D_; ++d) acc[g][d] = 0.0f;
    float wsum = 0.0f;

    const float xf = (float)w, yf = (float)h;

    for (int vi = 0; vi < V_ - 1; ++vi) {
        // proj_v = rows 4*vi..4*vi+2 of the WMMA dump
        const float* P = projws + b * 256 + vi * 64;
        const float r00 = P[0],  r01 = P[1],  r02 = P[2],  t0 = P[3];
        const float r10 = P[16], r11 = P[17], r12 = P[18], t1 = P[19];
        const float r20 = P[32], r21 = P[33], r22 = P[34], t2 = P[35];

        const float vw = vwts[((long)b * (V_ - 1) + vi) * HW_ + hw];
        wsum += vw;
        const float vq = 0.25f * vw;   // fold mean over 4 channels/group

        const float rx = r00 * xf + r01 * yf + r02;
        const float ry = r10 * xf + r11 * yf + r12;
        const float rz = r20 * xf + r21 * yf + r22;

        int   off[D_][4];
        float wgt[D_][4];
        #pragma unroll
        for (int d = 0; d < D_; ++d) {
            const float pz = rz * depth[d] + t2;
            const float gx = (rx * depth[d] + t0) / pz;
            const float gy = (ry * depth[d] + t1) / pz;
            const float x0f = floorf(gx), y0f = floorf(gy);
            const float fx = gx - x0f, fy = gy - y0f;
            const float x1f = x0f + 1.0f, y1f = y0f + 1.0f;
            const float vx0 = (x0f >= 0.0f && x0f <= (float)(W_-1)) ? 1.0f : 0.0f;
            const float vx1 = (x1f >= 0.0f && x1f <= (float)(W_-1)) ? 1.0f : 0.0f;
            const float vy0 = (y0f >= 0.0f && y0f <= (float)(H_-1)) ? 1.0f : 0.0f;
            const float vy1 = (y1f >= 0.0f && y1f <= (float)(H_-1)) ? 1.0f : 0.0f;
            const int ix0 = (int)fminf(fmaxf(x0f, 0.0f), (float)(W_-1));
            const int ix1 = (int)fminf(fmaxf(x1f, 0.0f), (float)(W_-1));
            const int iy0 = (int)fminf(fmaxf(y0f, 0.0f), (float)(H_-1));
            const int iy1 = (int)fminf(fmaxf(y1f, 0.0f), (float)(H_-1));
            off[d][0] = iy0 * W_ + ix0;
            off[d][1] = iy0 * W_ + ix1;
            off[d][2] = iy1 * W_ + ix0;
            off[d][3] = iy1 * W_ + ix1;
            wgt[d][0] = (1.0f - fx) * (1.0f - fy) * (vx0 * vy0);
            wgt[d][1] = fx * (1.0f - fy) * (vx1 * vy0);
            wgt[d][2] = (1.0f - fx) * fy * (vx0 * vy1);
            wgt[d][3] = fx * fy * (vx1 * vy1);
        }

        if (TR) {
            const float* sb = feat + ((long)(b * V_ + vi + 1) * HW_) * C_;
            #pragma unroll
            for (int d = 0; d < D_; ++d) {
                const float4* p0 = (const float4*)(sb + (long)off[d][0] * C_);
                const float4* p1 = (const float4*)(sb + (long)off[d][1] * C_);
                const float4* p2 = (const float4*)(sb + (long)off[d][2] * C_);
                const float4* p3 = (const float4*)(sb + (long)off[d][3] * C_);
                const float w0 = wgt[d][0], w1 = wgt[d][1];
                const float w2 = wgt[d][2], w3 = wgt[d][3];
                #pragma unroll
                for (int g = 0; g < G_; ++g) {
                    float4 c0 = p0[g], c1 = p1[g], c2 = p2[g], c3 = p3[g];
                    float sx = w0*c0.x + w1*c1.x + w2*c2.x + w3*c3.x;
                    float sy = w0*c0.y + w1*c1.y + w2*c2.y + w3*c3.y;
                    float sz = w0*c0.z + w1*c1.z + w2*c2.z + w3*c3.z;
                    float sw = w0*c0.w + w1*c1.w + w2*c2.w + w3*c3.w;
                    float dot = sx*refa[4*g+0] + sy*refa[4*g+1]
                              + sz*refa[4*g+2] + sw*refa[4*g+3];
                    acc[g][d] = fmaf(dot, vq, acc[g][d]);
                }
            }
        } else {
            const float* sb = feat + (long)(b * V_ + vi + 1) * C_ * HW_;
            #pragma unroll
            for (int g = 0; g < G_; ++g) {
                #pragma unroll
                for (int q = 0; q < 4; ++q) {
                    const float* cp = sb + (long)(4*g + q) * HW_;
                    const float rv = refa[4*g + q];
                    #pragma unroll
                    for (int d = 0; d < D_; ++d) {
                        float s = wgt[d][0] * cp[off[d][0]]
                                + wgt[d][1] * cp[off[d][1]]
                                + wgt[d][2] * cp[off[d][2]]
                                + wgt[d][3] * cp[off[d][3]];
                        acc[g][d] = fmaf(s * rv, vq, acc[g][d]);
                    }
                }
            }
        }
    }

    const float inv_wsum = 1.0f / (wsum + 1e-6f);
    #pragma unroll
    for (int g = 0; g < G_; ++g)
        #pragma unroll
        for (int d = 0; d < D_; ++d)
            simOut[((long)b * (G_ * D_) + g * D_ + d) * HW_ + hw] = acc[g][d] * inv_wsum;
}

// ---------------------------------------------------------------------------
extern "C" void kernel_launch(void* const* d_in, const int* in_sizes, int n_in,
                              void* d_out, int out_size, void* d_ws, size_t ws_size,
                              hipStream_t stream)
{
    (void)in_sizes; (void)n_in; (void)out_size;
    const float* depth_values   = (const float*)d_in[0];
    const float* features       = (const float*)d_in[1];
    const float* proj_matrices  = (const float*)d_in[2];
    const float* depth_interval = (const float*)d_in[3];
    // d_in[4]=depth_max, d_in[5]=depth_min unused
    const float* view_weights   = (const float*)d_in[6];
    // d_in[7]=CostNum(4), d_in[8]=G(8) baked in as compile-time constants

    float* simOut  = (float*)d_out;
    float* sampOut = simOut + (long)B_ * G_ * D_ * HW_;

    float* projws = (float*)d_ws;                 // B*256 floats (16x16 per batch)
    float* featT  = projws + 1024;                // 4 KB in
    const size_t needT = 4096 + (size_t)B_ * V_ * HW_ * C_ * sizeof(float);

    proj_wmma_kernel<<<dim3(B_), dim3(32), 0, stream>>>(proj_matrices, projws);

    const int npix = B_ * HW_;
    if (ws_size >= needT) {
        const int nt = B_ * V_ * HW_;
        transpose_kernel<<<(nt + 255) / 256, 256, 0, stream>>>(features, featT);
        getcost_kernel<true><<<(npix + 255) / 256, 256, 0, stream>>>(
            depth_values, depth_interval, view_weights, featT, projws, simOut, sampOut);
    } else {
        getcost_kernel<false><<<(npix + 255) / 256, 256, 0, stream>>>(
            depth_values, depth_interval, view_weights, features, projws, simOut, sampOut);
    }
}